// DirectSolverNet_20744692040061
// MI455X (gfx1250) — compile-verified
//
#include <hip/hip_runtime.h>
#include <cstdint>

// Problem constants (from reference): B=64, C=8, H=120, W=160
constexpr int BB     = 64;
constexpr int NN     = 8 * 120 * 160;   // 153600 elements per (b, row)
constexpr int SCHUNK = 50;              // chunks per batch
constexpr int STAGES = 3;               // pipeline stages per block (50*3*1024 = 153600)
constexpr int TPB    = 256;             // 8 wave32 waves
constexpr int WAVES  = 8;
constexpr int STREAMS = 8;              // 6 Jt rows + weights + R
constexpr int NITER  = 30;              // Newton-Schulz iterations (quadratic conv.)

typedef float v2f __attribute__((ext_vector_type(2)));
typedef float v8f __attribute__((ext_vector_type(8)));

// ---------------------------------------------------------------------------
// Kernel 1: JtR partial sums via async global->LDS double-buffered streaming.
// All streams are touched exactly once -> non-temporal hint keeps L2 clean.
// ---------------------------------------------------------------------------
__global__ __launch_bounds__(TPB) void jtr_partial_kernel(
    const float* __restrict__ Jt,
    const float* __restrict__ Wt,
    const float* __restrict__ Rr,
    float* __restrict__ partial)
{
  // [parity][wave][stream][lane] : 2*8*8*32*16B = 64 KB
  __shared__ float4 buf[2][WAVES][STREAMS][32];
  __shared__ float  red[WAVES][6];

  const int b    = blockIdx.y;
  const int s    = blockIdx.x;
  const int tid  = (int)threadIdx.x;
  const int wave = tid >> 5;
  const int lane = tid & 31;

  const size_t batchOff = (size_t)b * NN;
  const size_t jtBatch  = (size_t)b * 6 * NN;

  auto issue_stage = [&](int stage, int par) {
    const int n0 = s * (STAGES * 1024) + stage * 1024 + wave * 128 + lane * 4;
#pragma unroll
    for (int k = 0; k < 6; ++k) {
      const float* g = Jt + jtBatch + (size_t)k * NN + n0;
      uint32_t lo = (uint32_t)(uintptr_t)&buf[par][wave][k][lane];
      asm volatile("global_load_async_to_lds_b128 %0, %1, off th:TH_LOAD_NT"
                   :: "v"(lo), "v"(g) : "memory");
    }
    {
      const float* g = Wt + batchOff + n0;
      uint32_t lo = (uint32_t)(uintptr_t)&buf[par][wave][6][lane];
      asm volatile("global_load_async_to_lds_b128 %0, %1, off th:TH_LOAD_NT"
                   :: "v"(lo), "v"(g) : "memory");
    }
    {
      const float* g = Rr + batchOff + n0;
      uint32_t lo = (uint32_t)(uintptr_t)&buf[par][wave][7][lane];
      asm volatile("global_load_async_to_lds_b128 %0, %1, off th:TH_LOAD_NT"
                   :: "v"(lo), "v"(g) : "memory");
    }
  };

  float acc[6] = {0.f, 0.f, 0.f, 0.f, 0.f, 0.f};

  issue_stage(0, 0);                       // prologue: 8 async loads in flight
  for (int st = 0; st < STAGES; ++st) {
    const int par = st & 1;
    if (st + 1 < STAGES) {
      issue_stage(st + 1, (st + 1) & 1);   // 16 outstanding
      asm volatile("s_wait_asynccnt 8" ::: "memory");  // oldest stage landed
    } else {
      asm volatile("s_wait_asynccnt 0" ::: "memory");  // drain
    }
    const float4 w4 = buf[par][wave][6][lane];
    const float4 r4 = buf[par][wave][7][lane];
    const float wrx = w4.x * r4.x, wry = w4.y * r4.y;
    const float wrz = w4.z * r4.z, wrw = w4.w * r4.w;
#pragma unroll
    for (int k = 0; k < 6; ++k) {
      const float4 j4 = buf[par][wave][k][lane];
      acc[k] = fmaf(j4.x, wrx, acc[k]);
      acc[k] = fmaf(j4.y, wry, acc[k]);
      acc[k] = fmaf(j4.z, wrz, acc[k]);
      acc[k] = fmaf(j4.w, wrw, acc[k]);
    }
  }

  // wave32 reduction
#pragma unroll
  for (int off = 16; off > 0; off >>= 1) {
#pragma unroll
    for (int k = 0; k < 6; ++k) acc[k] += __shfl_down(acc[k], off, 32);
  }
  if (lane == 0) {
#pragma unroll
    for (int k = 0; k < 6; ++k) red[wave][k] = acc[k];
  }
  __syncthreads();
  if (tid < 6) {
    float ssum = 0.f;
#pragma unroll
    for (int w = 0; w < WAVES; ++w) ssum += red[w][tid];
    partial[((size_t)b * SCHUNK + s) * 6 + tid] = ssum;
  }
}

// ---------------------------------------------------------------------------
// WMMA helpers (V_WMMA_F32_16X16X4_F32, wave32, 16x16 operands padded in LDS)
// A fragment (ISA 7.12.2): lane<16 holds row M=lane, K={4q,4q+1}; lane>=16
// holds row M=lane-16, K={4q+2,4q+3}. B mirrors with N=lane. D: 8 VGPRs,
// element [M = r + (lane<16?0:8)][N = lane&15].
// ---------------------------------------------------------------------------
__device__ inline v8f wmma_mul16(const float (*Lm)[16], const float (*Rm)[16],
                                 int lane) {
  v8f c = {0.f, 0.f, 0.f, 0.f, 0.f, 0.f, 0.f, 0.f};
  const int half2 = (lane >> 4) * 2;
  const int l15   = lane & 15;
#pragma unroll
  for (int q = 0; q < 4; ++q) {
    v2f a, bfrag;
    a[0]     = Lm[l15][4 * q + half2 + 0];
    a[1]     = Lm[l15][4 * q + half2 + 1];
    bfrag[0] = Rm[4 * q + half2 + 0][l15];
    bfrag[1] = Rm[4 * q + half2 + 1][l15];
    c = __builtin_amdgcn_wmma_f32_16x16x4_f32(false, a, false, bfrag,
                                              (short)0, c, false, false);
  }
  return c;
}

__device__ inline void wmma_store16(float (*Out)[16], v8f d, int lane) {
  const int mb = (lane >> 4) ? 8 : 0;
  const int n  = lane & 15;
#pragma unroll
  for (int r = 0; r < 8; ++r) Out[mb + r][n] = d[r];
}

// ---------------------------------------------------------------------------
// Kernel 2: one wave per batch. Reduce partials, build padded damped Hessian,
// WMMA Newton-Schulz inverse X -> H^-1, xi = X * JtR, Rodrigues + pose tail.
// ---------------------------------------------------------------------------
__global__ __launch_bounds__(32) void solve_kernel(
    const float* __restrict__ JtJ,
    const float* __restrict__ partial,
    const float* __restrict__ poseR,
    const float* __restrict__ poseT,
    float* __restrict__ out)
{
  const int b    = blockIdx.x;
  const int lane = (int)threadIdx.x;   // one wave32

  __shared__ float Hs[16][16];
  __shared__ float Xs[16][16];
  __shared__ float Ys[16][16];
  __shared__ float jtr6[6];
  __shared__ float xiS[6];
  __shared__ float alphaS;

  // Reduce the 50 chunk partials
  if (lane < 6) {
    float ssum = 0.f;
    for (int s = 0; s < SCHUNK; ++s)
      ssum += partial[((size_t)b * SCHUNK + s) * 6 + lane];
    jtr6[lane] = ssum;
  }

  // Build padded H~ = blockdiag(JtJ, I10); each lane fills 8 entries
  for (int idx = lane; idx < 256; idx += 32) {
    const int i = idx >> 4, j = idx & 15;
    float v = (i == j) ? 1.f : 0.f;
    if (i < 6 && j < 6) v = JtJ[b * 36 + i * 6 + j];
    Hs[i][j] = v;
  }
  __syncthreads();

  // Damping + scaled-transpose init X0 = H~^T / (||H~||_1 * ||H~||_inf)
  if (lane == 0) {
    float tr = 0.f;
    for (int i = 0; i < 6; ++i) tr += Hs[i][i];
    for (int i = 0; i < 6; ++i) Hs[i][i] += tr * 1e-6f;
    float n1 = 0.f, ninf = 0.f;
    for (int i = 0; i < 16; ++i) {
      float rs = 0.f, cs = 0.f;
      for (int j = 0; j < 16; ++j) { rs += fabsf(Hs[i][j]); cs += fabsf(Hs[j][i]); }
      ninf = fmaxf(ninf, rs);
      n1   = fmaxf(n1, cs);
    }
    alphaS = 1.f / (n1 * ninf);
  }
  __syncthreads();

  const float alpha = alphaS;
  for (int idx = lane; idx < 256; idx += 32) {
    const int i = idx >> 4, j = idx & 15;
    Xs[i][j] = alpha * Hs[j][i];
  }
  __syncthreads();

  // Newton-Schulz: X <- X (2I - H X). H SPD (lam_min >= ~1 after padding),
  // quadratic convergence; NITER=30 is far past fp32 saturation.
  for (int it = 0; it < NITER; ++it) {
    v8f t = wmma_mul16(Hs, Xs, lane);            // T = H X
    const int mb = (lane >> 4) ? 8 : 0;
    const int n  = lane & 15;
    v8f yv;
#pragma unroll
    for (int r = 0; r < 8; ++r) {
      const int M = mb + r;
      yv[r] = ((M == n) ? 2.f : 0.f) - t[r];     // Y = 2I - T
    }
    wmma_store16(Ys, yv, lane);
    __syncthreads();
    v8f xn = wmma_mul16(Xs, Ys, lane);           // X' = X Y
    __syncthreads();                             // all reads of Xs done
    wmma_store16(Xs, xn, lane);
    __syncthreads();
  }

  // xi = H^-1 * JtR  (top-left 6x6 block of X)
  if (lane < 6) {
    float ssum = 0.f;
    for (int j = 0; j < 6; ++j) ssum += Xs[lane][j] * jtr6[j];
    xiS[lane] = ssum;
  }
  __syncthreads();
  if (lane != 0) return;

  float xi[6];
  for (int i = 0; i < 6; ++i) xi[i] = xiS[i];

  // dR = twist2mat(-xi[:3]) exactly as the reference computes it
  const float wx = -xi[0], wy = -xi[1], wz = -xi[2];
  float theta = sqrtf(wx * wx + wy * wy + wz * wz);
  theta = fmaxf(theta, 1e-12f);
  const float ux = wx / theta, uy = wy / theta, uz = wz / theta;
  const float sth = sinf(theta), cth = cosf(theta), vth = 1.f - cth;

  const float K[3][3] = {{0.f, -uz,  uy},
                         { uz, 0.f, -ux},
                         {-uy,  ux, 0.f}};
  float K2[3][3];
  for (int i = 0; i < 3; ++i)
    for (int j = 0; j < 3; ++j) {
      float sum = 0.f;
      for (int k = 0; k < 3; ++k) sum += K[i][k] * K[k][j];
      K2[i][j] = sum;
    }
  float dR[3][3];
  for (int i = 0; i < 3; ++i)
    for (int j = 0; j < 3; ++j)
      dR[i][j] = ((i == j) ? 1.f : 0.f) + sth * K[i][j] + vth * K2[i][j];

  // dt = -(dR @ xi[3:])
  const float v3[3] = {xi[3], xi[4], xi[5]};
  float dt[3];
  for (int i = 0; i < 3; ++i)
    dt[i] = -(dR[i][0] * v3[0] + dR[i][1] * v3[1] + dR[i][2] * v3[2]);

  // R_new = pose_R @ dR ; t_new = pose_R @ dt + pose_t
  const float* P = poseR + b * 9;
  float Rn[3][3], tn[3];
  for (int i = 0; i < 3; ++i) {
    for (int j = 0; j < 3; ++j) {
      float sum = 0.f;
      for (int k = 0; k < 3; ++k) sum += P[i * 3 + k] * dR[k][j];
      Rn[i][j] = sum;
    }
    tn[i] = P[i * 3 + 0] * dt[0] + P[i * 3 + 1] * dt[1] + P[i * 3 + 2] * dt[2]
          + poseT[b * 3 + i];
  }

  // Output: R_new (B,3,3) flat, then t_new (B,3) flat
  for (int i = 0; i < 3; ++i)
    for (int j = 0; j < 3; ++j)
      out[b * 9 + i * 3 + j] = Rn[i][j];
  for (int i = 0; i < 3; ++i)
    out[BB * 9 + b * 3 + i] = tn[i];
}

// ---------------------------------------------------------------------------
extern "C" void kernel_launch(void* const* d_in, const int* in_sizes, int n_in,
                              void* d_out, int out_size, void* d_ws, size_t ws_size,
                              hipStream_t stream) {
  (void)in_sizes; (void)n_in; (void)out_size; (void)ws_size;
  const float* JtJ   = (const float*)d_in[0];
  const float* Jt    = (const float*)d_in[1];
  const float* Wt    = (const float*)d_in[2];
  const float* Rr    = (const float*)d_in[3];
  const float* poseR = (const float*)d_in[4];
  const float* poseT = (const float*)d_in[5];
  // invD0/invD1/x0/x1/K (d_in[6..10]) are unused by the reference.

  float* partial = (float*)d_ws;  // 64*50*6 floats = 76.8 KB, fully rewritten every call

  dim3 g1(SCHUNK, BB), b1(TPB);
  hipLaunchKernelGGL(jtr_partial_kernel, g1, b1, 0, stream, Jt, Wt, Rr, partial);
  hipLaunchKernelGGL(solve_kernel, dim3(BB), dim3(32), 0, stream,
                     JtJ, partial, poseR, poseT, (float*)d_out);
}